// Net_37512244363273
// MI455X (gfx1250) — compile-verified
//
#include <hip/hip_runtime.h>
#include <hip/hip_bf16.h>

typedef __attribute__((ext_vector_type(2))) float v2f;
typedef __attribute__((ext_vector_type(8))) float v8f;

#define KK 5
#define KEY_EMPTY ((int)0x80000000)

__device__ __forceinline__ int fkey(float f) {
    int b = __float_as_int(f);
    return b ^ ((b >> 31) & 0x7fffffff);
}
__device__ __forceinline__ float fdec(int k) {
    int b = k ^ ((k >> 31) & 0x7fffffff);
    return __int_as_float(b);
}

// ---------------------------------------------------------------- utilities
__global__ void fill_i32(int* __restrict__ p, int v, int n) {
    int i = blockIdx.x * blockDim.x + threadIdx.x;
    if (i < n) p[i] = v;
}

// ------------------------------------------------ per-level prep (1 block)
// Gbuf[0..24]  = G[k1][k2] = sum_io w2[k1,io]*w2[k2,io]
// Gbuf[25..29] = g1[k]     = sum_io w2[k,io]*b2[io]
// Gbuf[30]     = sum_io b2[io]^2
// W2p[(i*8+k)*co_pad + o] = w2[k, i*co+o]   (k<5, o<co; zero padded)
__global__ void prep_level(const float* __restrict__ w2, const float* __restrict__ b2,
                           float* __restrict__ Gbuf, float* __restrict__ W2p,
                           int ci, int co, int co_pad) {
    int t = threadIdx.x;
    int cico = ci * co;
    if (t < 25) {
        int k1 = t / 5, k2 = t % 5;
        float s = 0.f;
        for (int j = 0; j < cico; ++j) s += w2[k1 * cico + j] * w2[k2 * cico + j];
        Gbuf[t] = s;
    } else if (t < 30) {
        int k = t - 25;
        float s = 0.f;
        for (int j = 0; j < cico; ++j) s += w2[k * cico + j] * b2[j];
        Gbuf[t] = s;
    } else if (t == 30) {
        float s = 0.f;
        for (int j = 0; j < cico; ++j) s += b2[j] * b2[j];
        Gbuf[30] = s;
    }
    int total = ci * 8 * co_pad;
    for (int idx = t; idx < total; idx += blockDim.x) {
        int row = idx / co_pad, o = idx % co_pad;
        int i = row >> 3, k = row & 7;
        float v = 0.f;
        if (k < KK && o < co) v = w2[k * cico + i * co + o];
        W2p[idx] = v;
    }
}

// ---------------------------------------------- edge conv (WMMA f32 16x16x4)
// Each wave handles 16 edges. Z[e, i*8+k] = x[src_e, i] * h[e, k] (k padded to 8).
// msg(16 x co) = Z(16 x 8ci) @ W2p(8ci x co_pad); accumulate over K in steps of 4.
__global__ void conv_edges(const float* __restrict__ x, const int* __restrict__ src,
                           const int* __restrict__ dst, const float* __restrict__ eattr,
                           const float* __restrict__ w1, const float* __restrict__ b1,
                           const float* __restrict__ G, const float* __restrict__ W2p,
                           float* __restrict__ agg, float* __restrict__ cnt,
                           float* __restrict__ closs, int E, int ci, int co, int co_pad) {
    __shared__ float hs[8][16][8];
    __shared__ float xs[8][16][40];
    __shared__ int   ds[8][16];

    const int tid  = threadIdx.x;
    const int wave = tid >> 5;
    const int lane = tid & 31;
    const int tile = blockIdx.x * 8 + wave;
    const int e0   = tile * 16;

    float q = 0.0f;
    if (lane < 16) {
        int e = e0 + lane;
        bool valid = (e < E);
        float h[KK];
        if (valid) {
            float a0 = eattr[e * 3 + 0], a1 = eattr[e * 3 + 1], a2 = eattr[e * 3 + 2];
#pragma unroll
            for (int k = 0; k < KK; ++k)
                h[k] = tanhf(a0 * w1[0 * KK + k] + a1 * w1[1 * KK + k] +
                             a2 * w1[2 * KK + k] + b1[k]);
            // closs: ||h*w2 + b2||^2 = h^T G h + 2 h.g1 + ||b2||^2
#pragma unroll
            for (int k1 = 0; k1 < KK; ++k1) {
                float s = 0.f;
#pragma unroll
                for (int k2 = 0; k2 < KK; ++k2) s += G[k1 * KK + k2] * h[k2];
                q += h[k1] * s + 2.0f * G[25 + k1] * h[k1];
            }
            q += G[30];
            int sv = src[e];
            for (int i = 0; i < ci; ++i) xs[wave][lane][i] = x[sv * ci + i];
            ds[wave][lane] = dst[e];
            atomicAdd(&cnt[dst[e]], 1.0f);
        } else {
#pragma unroll
            for (int k = 0; k < KK; ++k) h[k] = 0.f;
            for (int i = 0; i < ci; ++i) xs[wave][lane][i] = 0.f;
            ds[wave][lane] = 0;
        }
#pragma unroll
        for (int k = 0; k < KK; ++k) hs[wave][lane][k] = h[k];
#pragma unroll
        for (int k = KK; k < 8; ++k) hs[wave][lane][k] = 0.f;
    }
    __syncthreads();

    // wave-reduce closs contribution (q==0 on lanes 16..31)
    for (int off = 16; off > 0; off >>= 1) q += __shfl_down(q, off, 32);
    if (lane == 0) atomicAdd(closs, q);

    const int m    = lane & 15;   // A row / C column
    const int hi   = lane >> 4;   // lane-half
    const int Kdim = ci * 8;
    const int ntiles = (co + 15) >> 4;

    for (int ct = 0; ct < ntiles; ++ct) {
        v8f acc = {};
        const int col = ct * 16 + m;
        for (int kd0 = 0; kd0 < Kdim; kd0 += 4) {
            // A 16x4 f32: lanes 0-15 hold K={0,1}, lanes 16-31 hold K={2,3}
            int ka = kd0 + hi * 2;
            v2f a, b;
            a.x = xs[wave][m][ka >> 3]       * hs[wave][m][ka & 7];
            a.y = xs[wave][m][(ka + 1) >> 3] * hs[wave][m][(ka + 1) & 7];
            // B 4x16 f32: lanes 0-15 hold rows {0,1}, lanes 16-31 rows {2,3}
            b.x = W2p[(kd0 + hi * 2) * co_pad + col];
            b.y = W2p[(kd0 + hi * 2 + 1) * co_pad + col];
            acc = __builtin_amdgcn_wmma_f32_16x16x4_f32(
                false, a, false, b, (short)0, acc, false, false);
        }
        if (col < co) {
#pragma unroll
            for (int r = 0; r < 8; ++r) {
                int row = hi * 8 + r;       // C: lanes 0-15 rows 0-7, lanes 16-31 rows 8-15
                int e = e0 + row;
                if (e < E) atomicAdd(&agg[ds[wave][row] * co + col], acc[r]);
            }
        }
    }
}

// -------------------------------------- node update: out = agg/cnt + x@root + b
__global__ void node_update(const float* __restrict__ x, const float* __restrict__ root,
                            const float* __restrict__ bias, const float* __restrict__ cnt,
                            float* __restrict__ xb, int n, int ci, int co) {
    int idx = blockIdx.x * blockDim.x + threadIdx.x;
    if (idx >= n * co) return;
    int v = idx / co, o = idx % co;
    float c = fmaxf(cnt[v], 1.0f);
    float r = 0.f;
    for (int i = 0; i < ci; ++i) r += x[v * ci + i] * root[i * co + o];
    xb[idx] = xb[idx] / c + r + bias[o];
}

// ------------------------------------------------------------- pooling
__global__ void pool_max(const float* __restrict__ xb, const int* __restrict__ cl,
                         int* __restrict__ keys, int n, int co) {
    int idx = blockIdx.x * blockDim.x + threadIdx.x;
    if (idx >= n * co) return;
    int v = idx / co, o = idx % co;
    atomicMax(&keys[cl[v] * co + o], fkey(xb[idx]));
}

__global__ void pool_pos(const float* __restrict__ pos, const int* __restrict__ cl,
                         float* __restrict__ psum, float* __restrict__ pcnt, int n) {
    int v = blockIdx.x * blockDim.x + threadIdx.x;
    if (v >= n) return;
    int c = cl[v];
    atomicAdd(&psum[c * 3 + 0], pos[v * 3 + 0]);
    atomicAdd(&psum[c * 3 + 1], pos[v * 3 + 1]);
    atomicAdd(&psum[c * 3 + 2], pos[v * 3 + 2]);
    atomicAdd(&pcnt[c], 1.0f);
}

__global__ void pool_finalize(const int* __restrict__ keys, float* __restrict__ psum,
                              const float* __restrict__ pcnt, float* __restrict__ xnext,
                              int n2, int co) {
    int c = blockIdx.x * blockDim.x + threadIdx.x;
    if (c >= n2) return;
    float cc  = fmaxf(pcnt[c], 1.0f);
    float pm0 = psum[c * 3 + 0] / cc;
    float pm1 = psum[c * 3 + 1] / cc;
    float pm2 = psum[c * 3 + 2] / cc;
    psum[c * 3 + 0] = pm0; psum[c * 3 + 1] = pm1; psum[c * 3 + 2] = pm2; // pos_next in place
    int stride = co + 3;
    for (int o = 0; o < co; ++o) {
        int k = keys[c * co + o];
        float f = 0.f;
        if (k != KEY_EMPTY) {
            f = fdec(k);
            if (!isfinite(f)) f = 0.f;
        }
        xnext[c * stride + o] = f;
    }
    xnext[c * stride + co + 0] = pm0;
    xnext[c * stride + co + 1] = pm1;
    xnext[c * stride + co + 2] = pm2;
}

// --------------------------------------------- head: FC + log_softmax + closs
__global__ void head_kernel(const float* __restrict__ x, const float* __restrict__ fcw,
                            const float* __restrict__ fcb, const float* __restrict__ closs5,
                            float* __restrict__ out) {
    __shared__ float lg[8][10];
    __shared__ float rmax[8], rlse[8];
    int t = threadIdx.x;
    if (t < 80) {
        int b = t / 10, j = t % 10;
        float s = fcb[j];
        for (int f = 0; f < 376; ++f) s += x[b * 376 + f] * fcw[f * 10 + j];
        lg[b][j] = s;
    }
    __syncthreads();
    if (t < 8) {
        float mx = -1e30f;
        for (int j = 0; j < 10; ++j) mx = fmaxf(mx, lg[t][j]);
        float sum = 0.f;
        for (int j = 0; j < 10; ++j) sum += __expf(lg[t][j] - mx);
        rmax[t] = mx; rlse[t] = logf(sum);
    }
    __syncthreads();
    if (t < 80) {
        int b = t / 10, j = t % 10;
        out[t] = lg[b][j] - rmax[b] - rlse[b];
    }
    if (t == 0) {
        const float denom[5] = {12582912.f, 78643200.f, 42205184.f, 18284544.f, 7028736.f};
        float c = 0.f;
        for (int l = 0; l < 5; ++l) c += closs5[l] / denom[l];
        out[80] = c;
    }
}

// ---------------------------------------------------------------- launcher
extern "C" void kernel_launch(void* const* d_in, const int* in_sizes, int n_in,
                              void* d_out, int out_size, void* d_ws, size_t ws_size,
                              hipStream_t stream) {
    (void)in_sizes; (void)n_in; (void)out_size; (void)ws_size;
    static const int NSh[6]  = {65536, 16384, 4096, 1024, 256, 64};
    static const int ESh[5]  = {1048576, 262144, 65536, 16384, 4096};
    static const int INSh[5] = {1, 15, 23, 31, 39};
    static const int OUTSh[5]= {12, 20, 28, 36, 44};

    float* ws = (float*)d_ws;
    float* xPing   = ws;                 // 262144
    float* xPong   = ws + 262144;        // 262144
    float* aggB    = ws + 524288;        // 786432 (max 65536*12)
    float* posPing = ws + 1310720;       // 196608
    float* posPong = ws + 1507328;       // 196608
    float* cntB    = ws + 1703936;       // 65536
    int*   keyB    = (int*)(ws + 1769472); // 262144
    float* W2p     = ws + 2031616;       // 16384 (max 312*48)
    float* Gbuf    = ws + 2048000;       // 64
    float* closs5  = ws + 2048064;       // 8

    hipMemsetAsync(closs5, 0, 5 * sizeof(float), stream);

    const float* xcur   = (const float*)d_in[0];
    const float* poscur = (const float*)d_in[1];

    for (int l = 0; l < 5; ++l) {
        const int base = 2 + l * 10;
        const int*   src  = (const int*)  d_in[base + 0];
        const int*   dst  = (const int*)  d_in[base + 1];
        const float* ea   = (const float*)d_in[base + 2];
        const int*   clu  = (const int*)  d_in[base + 3];
        const float* w1   = (const float*)d_in[base + 4];
        const float* b1   = (const float*)d_in[base + 5];
        const float* w2   = (const float*)d_in[base + 6];
        const float* b2   = (const float*)d_in[base + 7];
        const float* root = (const float*)d_in[base + 8];
        const float* bias = (const float*)d_in[base + 9];

        const int n = NSh[l], E = ESh[l], ci = INSh[l], co = OUTSh[l], n2 = NSh[l + 1];
        const int co_pad = ((co + 15) / 16) * 16;

        hipMemsetAsync(aggB, 0, (size_t)n * co * sizeof(float), stream);
        hipMemsetAsync(cntB, 0, (size_t)n * sizeof(float), stream);
        prep_level<<<1, 256, 0, stream>>>(w2, b2, Gbuf, W2p, ci, co, co_pad);

        const int tiles  = (E + 15) / 16;
        const int blocks = (tiles + 7) / 8;
        conv_edges<<<blocks, 256, 0, stream>>>(xcur, src, dst, ea, w1, b1, Gbuf, W2p,
                                               aggB, cntB, closs5 + l, E, ci, co, co_pad);

        node_update<<<(n * co + 255) / 256, 256, 0, stream>>>(xcur, root, bias, cntB,
                                                              aggB, n, ci, co);

        float* psum  = (l & 1) ? posPong : posPing;
        float* xnext = (l & 1) ? xPong : xPing;

        fill_i32<<<(n2 * co + 255) / 256, 256, 0, stream>>>(keyB, KEY_EMPTY, n2 * co);
        hipMemsetAsync(psum, 0, (size_t)n2 * 3 * sizeof(float), stream);
        hipMemsetAsync(cntB, 0, (size_t)n2 * sizeof(float), stream);

        pool_max<<<(n * co + 255) / 256, 256, 0, stream>>>(aggB, clu, keyB, n, co);
        pool_pos<<<(n + 255) / 256, 256, 0, stream>>>(poscur, clu, psum, cntB, n);
        pool_finalize<<<(n2 + 255) / 256, 256, 0, stream>>>(keyB, psum, cntB, xnext, n2, co);

        xcur = xnext;
        poscur = psum;
    }

    const float* fcw = (const float*)d_in[52];
    const float* fcb = (const float*)d_in[53];
    head_kernel<<<1, 128, 0, stream>>>(xcur, fcw, fcb, closs5, (float*)d_out);
}